// graph_conv_with_jk_28991029248104
// MI455X (gfx1250) — compile-verified
//
#include <hip/hip_runtime.h>
#include <hip/hip_bf16.h>

// GraphConv-with-Chebyshev fused kernel for gfx1250 (MI455X).
// out[b,t] = relu( sum_k T_k(b)^T @ x[b,t] @ Theta_k ) + sigmoid(x[b,t])
// T_0 = I, T_1 = Lt, T_k = 2*Lt.*T_{k-1} - T_{k-2} (ELEMENTWISE recurrence),
// Lt = diag(colsum(W)) - W - I.
//
// Memory-bound: 256 MB x in + 256 MB out => ~22us floor @ 23.3 TB/s.
// Matmuls: V_WMMA_F32_16X16X4_F32. x tiles stream into LDS via the Tensor
// Data Mover (tensor_load_to_lds, TENSORcnt) with double buffering so the
// next timestep's 16 KB tile is in flight while the current one computes.

typedef float v2f __attribute__((ext_vector_type(2)));
typedef float v8f __attribute__((ext_vector_type(8)));
typedef unsigned int u32x4 __attribute__((ext_vector_type(4)));
typedef int i32x4 __attribute__((ext_vector_type(4)));
typedef int i32x8 __attribute__((ext_vector_type(8)));

#define B_DIM 32
#define T_DIM 512
#define V_DIM 64
#define TT_PER_BLOCK 16

#if __has_builtin(__builtin_amdgcn_tensor_load_to_lds) && \
    __has_builtin(__builtin_amdgcn_s_wait_tensorcnt)
#define USE_TDM 1
#else
#define USE_TDM 0
#endif

#if USE_TDM
// Byte offset of a __shared__ object inside the workgroup LDS allocation.
__device__ __forceinline__ unsigned lds_offset_of(const void* p) {
  return (unsigned)(unsigned long long)(__attribute__((address_space(3))) const char*)p;
}

// Issue a TDM load of one 64x64 f32 tile (row-major, stride 64) from global
// memory into LDS at byte offset lds_off. Call from a single wave; completion
// tracked by that wave's TENSORcnt.
__device__ __forceinline__ void tdm_load_tile(const float* gsrc, unsigned lds_off) {
  const unsigned long long ga = (unsigned long long)(uintptr_t)gsrc;
  u32x4 g0;
  g0.x = 1u;                                   // count=1, user descriptor
  g0.y = lds_off;                              // lds_addr (bytes)
  g0.z = (unsigned)(ga & 0xFFFFFFFFu);         // global_addr[31:0]
  g0.w = (unsigned)((ga >> 32) & 0x01FFFFFFu)  // global_addr[56:32]
         | (2u << 30);                         // type=2 ("image")
  i32x8 g1;
  g1[0] = (int)(2u << 16);   // workgroup_mask=0, data_size=2 (4 bytes)
  g1[1] = (int)(64u << 16);  // atomic_barrier_addr=0 | tensor_dim0.lo16 = 64
  g1[2] = (int)(64u << 16);  // tensor_dim0.hi16 = 0  | tensor_dim1.lo16 = 64
  g1[3] = (int)(64u << 16);  // tensor_dim1.hi16 = 0  | tile_dim0 = 64
  g1[4] = 64;                // tile_dim1 = 64, tile_dim2 = 0
  g1[5] = 64;                // tensor_dim0_stride.lo32 = 64 elements
  g1[6] = 0;                 // stride0.hi16 = 0, tensor_dim1_stride.lo16 = 0
  g1[7] = 0;
  const i32x4 z4 = {0, 0, 0, 0};               // groups 2/3 unused (2D tensor)
  const i32x8 z8 = {0, 0, 0, 0, 0, 0, 0, 0};   // extra group (6-arg toolchain)
  __builtin_amdgcn_tensor_load_to_lds(g0, g1, z4, z4, z8, 0);
}
#endif

// Accumulate a pair of 16x16 output tiles (rows i0*16.., i1*16.., cols j*16..)
// of D += A(64x64) @ B(64x64), A/B row-major in LDS. The B operand is shared
// between the two accumulators (i0 and i1 differ, j is the same).
__device__ __forceinline__ void mma_pair(const float* Amat, const float* Bmat,
                                         int i0, int i1, int j,
                                         int row16, int half,
                                         v8f& acc0, v8f& acc1) {
#pragma unroll
  for (int kk = 0; kk < 16; ++kk) {
    const int kb = kk * 4 + 2 * half;        // K base for this lane-half
    v2f bv;                                  // B: row K = vgpr + 2*half
    bv.x = Bmat[(kb + 0) * 64 + j * 16 + row16];
    bv.y = Bmat[(kb + 1) * 64 + j * 16 + row16];
    v2f a0 = *(const v2f*)&Amat[(i0 * 16 + row16) * 64 + kb];
    v2f a1 = *(const v2f*)&Amat[(i1 * 16 + row16) * 64 + kb];
    acc0 = __builtin_amdgcn_wmma_f32_16x16x4_f32(false, a0, false, bv,
                                                 (short)0, acc0, false, false);
    acc1 = __builtin_amdgcn_wmma_f32_16x16x4_f32(false, a1, false, bv,
                                                 (short)0, acc1, false, false);
  }
}

// Store a 16x16 f32 C/D tile to LDS (row-major 64x64). M = vgpr + 8*half.
__device__ __forceinline__ void store_tile(float* M, int i, int j,
                                           int row16, int half, const v8f& acc) {
#pragma unroll
  for (int r = 0; r < 8; ++r) {
    M[(i * 16 + r + 8 * half) * 64 + j * 16 + row16] = acc[r];
  }
}

__global__ __launch_bounds__(256) void graph_cheb_fused_kernel(
    const float* __restrict__ x,      // (B,T,V,F)
    const float* __restrict__ W,      // (B,V,V)
    const float* __restrict__ Theta,  // (4,F,F)
    float* __restrict__ out) {        // (B,T,V,F)
  __shared__ __align__(16) float sTheta[4 * 4096];  // 64 KB
  __shared__ __align__(16) float sTt[3 * 4096];     // 48 KB  (T_k^T, k=1..3)
#if USE_TDM
  __shared__ __align__(16) float sX[2][4096];       // 32 KB ping-pong
#else
  __shared__ __align__(16) float sX[1][4096];       // 16 KB
#endif
  __shared__ __align__(16) float sZ[4096];          // 16 KB (also temp deg[])
  float* sDeg = sZ;  // alias: deg[] only lives during setup, before sZ is used

  const int tid = threadIdx.x;
  const int b = blockIdx.x >> 5;
  const int tchunk = blockIdx.x & 31;
  const int t0 = tchunk * TT_PER_BLOCK;

#if USE_TDM
  // Kick off the DMA for the first timestep before doing any setup work.
  if (tid < 32) {
    tdm_load_tile(x + ((size_t)(b * T_DIM + t0)) * 4096, lds_offset_of(&sX[0][0]));
  }
#else
  float4 pre[4];
  {
    const float* xsrc = x + ((size_t)(b * T_DIM + t0)) * 4096;
#pragma unroll
    for (int r = 0; r < 4; ++r) pre[r] = ((const float4*)xsrc)[r * 256 + tid];
  }
#endif

  // ---- Load Theta into LDS (all k) ----
#pragma unroll 4
  for (int e = tid; e < 4 * 4096; e += 256) sTheta[e] = Theta[e];

  // ---- deg[c] = sum_u W[b,u,c]  (column sums) ----
  const float* Wb = W + (size_t)b * 4096;
  if (tid < 64) {
    float s = 0.f;
    for (int w = 0; w < 64; ++w) s += Wb[w * 64 + tid];
    sDeg[tid] = s;
  }
  __syncthreads();

  // ---- Chebyshev recurrence (elementwise), stored transposed ----
  for (int e = tid; e < 4096; e += 256) {
    const int u = e >> 6, v = e & 63;
    const float w_uv = Wb[e];
    const float diag = (u == v) ? 1.f : 0.f;
    const float lt = sDeg[u] * diag - w_uv - diag;   // (D - W - I)[u,v]
    const float t2 = 2.f * lt * lt - diag;
    const float t3 = 2.f * lt * t2 - lt;
    sTt[0 * 4096 + v * 64 + u] = lt;   // T_1^T
    sTt[1 * 4096 + v * 64 + u] = t2;   // T_2^T
    sTt[2 * 4096 + v * 64 + u] = t3;   // T_3^T
  }
#if USE_TDM
  if (tid < 32) __builtin_amdgcn_s_wait_tensorcnt(0);  // sX[0] landed
#endif
  __syncthreads();

  // ---- Wave/tile assignment ----
  const int wave = tid >> 5;
  const int lane = tid & 31;
  const int half = lane >> 4;
  const int row16 = lane & 15;
  const int i0 = wave >> 2;      // 0..1
  const int i1 = i0 + 2;         // 2..3  (shares column block j -> B reuse)
  const int j = wave & 3;        // 0..3

#pragma unroll 1
  for (int tt = 0; tt < TT_PER_BLOCK; ++tt) {
    const int t = t0 + tt;
#if USE_TDM
    const int cur = tt & 1;
    // Prefetch the next tile into the other buffer while this one computes.
    if (tt + 1 < TT_PER_BLOCK && tid < 32) {
      tdm_load_tile(x + ((size_t)(b * T_DIM + t + 1)) * 4096,
                    lds_offset_of(&sX[1 - cur][0]));
    }
    const float* xlds = sX[cur];
#else
    const int cur = 0;
    __syncthreads();  // previous iteration done reading sX
#pragma unroll
    for (int r = 0; r < 4; ++r) ((float4*)sX[0])[r * 256 + tid] = pre[r];
    __syncthreads();
    if (tt + 1 < TT_PER_BLOCK) {
      const float* xsrc = x + ((size_t)(b * T_DIM + t + 1)) * 4096;
#pragma unroll
      for (int r = 0; r < 4; ++r) pre[r] = ((const float4*)xsrc)[r * 256 + tid];
    }
    const float* xlds = sX[0];
#endif

    v8f acc0 = {}, acc1 = {};
    // k = 0: T_0 = I  =>  out += x @ Theta_0 directly.
    mma_pair(xlds, sTheta, i0, i1, j, row16, half, acc0, acc1);

#pragma unroll 1
    for (int k = 1; k < 4; ++k) {
      v8f z0 = {}, z1 = {};
      mma_pair(xlds, sTheta + k * 4096, i0, i1, j, row16, half, z0, z1);
      __syncthreads();  // everyone done reading sZ from previous k
      store_tile(sZ, i0, j, row16, half, z0);
      store_tile(sZ, i1, j, row16, half, z1);
      __syncthreads();  // sZ fully written
      mma_pair(sTt + (k - 1) * 4096, sZ, i0, i1, j, row16, half, acc0, acc1);
    }

    // ---- Epilogue: relu(acc) + sigmoid(x), non-temporal store ----
    float* outb = out + ((size_t)(b * T_DIM + t)) * 4096;
    const int g = j * 16 + row16;
#pragma unroll
    for (int r = 0; r < 8; ++r) {
      const int v0 = i0 * 16 + r + 8 * half;
      const int v1 = i1 * 16 + r + 8 * half;
      const float xv0 = xlds[v0 * 64 + g];
      const float xv1 = xlds[v1 * 64 + g];
      __builtin_nontemporal_store(
          fmaxf(acc0[r], 0.f) + 1.f / (1.f + __expf(-xv0)), &outb[v0 * 64 + g]);
      __builtin_nontemporal_store(
          fmaxf(acc1[r], 0.f) + 1.f / (1.f + __expf(-xv1)), &outb[v1 * 64 + g]);
    }

#if USE_TDM
    // Ensure the prefetched tile has landed and every wave is done with this
    // iteration's buffers before the next iteration reuses them.
    if (tid < 32) __builtin_amdgcn_s_wait_tensorcnt(0);
    __syncthreads();
#endif
  }
}

extern "C" void kernel_launch(void* const* d_in, const int* in_sizes, int n_in,
                              void* d_out, int out_size, void* d_ws, size_t ws_size,
                              hipStream_t stream) {
  const float* x = (const float*)d_in[0];      // (32,512,64,64)
  const float* W = (const float*)d_in[1];      // (32,64,64)
  const float* Theta = (const float*)d_in[2];  // (4,64,64)
  float* out = (float*)d_out;

  dim3 grid(B_DIM * (T_DIM / TT_PER_BLOCK));   // 1024 blocks
  dim3 block(256);                             // 8 waves (wave32)
  hipLaunchKernelGGL(graph_cheb_fused_kernel, grid, block, 0, stream,
                     x, W, Theta, out);
}